// ParametricEQ_56538949485206
// MI455X (gfx1250) — compile-verified
//
#include <hip/hip_runtime.h>
#include <math.h>

typedef __attribute__((ext_vector_type(2))) float v2f;
typedef __attribute__((ext_vector_type(8))) float v8f;

#define NBATCH 32
#define T_LEN 131072
#define BLK 16                     // samples per block
#define NBLOCKS (T_LEN / BLK)      // 8192
#define GRP 16                     // blocks per WMMA tile (N dim)
#define NGROUPS (NBLOCKS / GRP)    // 512
#define CHUNK_BLOCKS 256
#define NCHUNKS (NBLOCKS / CHUNK_BLOCKS)        // 32
#define GROUPS_PER_CHUNK (CHUNK_BLOCKS / GRP)   // 16
#define MATS_PER_BATCH 5

// ---- workspace layout (in floats) ----
// per-batch 16x16 matrices: Tm, Dm(16x10 pad), Bm(10x16 pad), Am(10x10 pad), Am256
#define MAT_OFF    ((size_t)0)
#define SX_OFF     (MAT_OFF + (size_t)NBATCH * MATS_PER_BATCH * 256)   // 40960
#define CHUNKF_OFF (SX_OFF + (size_t)NBATCH * NGROUPS * 256)           // +4194304
#define CHUNKS_OFF (CHUNKF_OFF + (size_t)NBATCH * NCHUNKS * 16)
#define WS_FLOATS  (CHUNKS_OFF + (size_t)NBATCH * NCHUNKS * 16)

__constant__ float kLo[15] = {-12.f,20.f,0.1f, -12.f,20.f,0.1f, -12.f,200.f,0.1f,
                              -12.f,2000.f,0.1f, -12.f,4000.f,0.1f};
__constant__ float kHi[15] = { 12.f,2000.f,10.f, 12.f,200.f,10.f, 12.f,2000.f,10.f,
                               12.f,12000.f,10.f, 12.f,16000.f,10.f};

static __device__ __forceinline__ v8f wmma4(v2f a, v2f b, v8f c) {
  return __builtin_amdgcn_wmma_f32_16x16x4_f32(false, a, false, b, (short)0, c, false, false);
}

// ---------------- Kernel 1: RBJ biquad SOS coefficients (second output) --------------
__global__ void eq_sos_kernel(const float* __restrict__ cp, float* __restrict__ sosOut) {
  int t = blockIdx.x * blockDim.x + threadIdx.x;
  if (t >= NBATCH * 5) return;
  int b = t / 5, sec = t % 5;
  const float* p = cp + b * 15 + sec * 3;
  float gain = kLo[sec*3+0] + p[0] * (kHi[sec*3+0] - kLo[sec*3+0]);
  float f0   = kLo[sec*3+1] + p[1] * (kHi[sec*3+1] - kLo[sec*3+1]);
  float q    = kLo[sec*3+2] + p[2] * (kHi[sec*3+2] - kLo[sec*3+2]);

  float A  = powf(10.0f, gain * (1.0f/40.0f));
  float w0 = 2.0f * 3.14159265358979323846f * (f0 / 44100.0f);
  float sw = sinf(w0), cw = cosf(w0);
  float alpha = sw / (2.0f * q);
  float sA = sqrtf(A);
  float b0,b1,b2,a0,a1,a2;
  if (sec == 0) {            // low shelf
    b0 =  A*((A+1.f) - (A-1.f)*cw + 2.f*sA*alpha);
    b1 =  2.f*A*((A-1.f) - (A+1.f)*cw);
    b2 =  A*((A+1.f) - (A-1.f)*cw - 2.f*sA*alpha);
    a0 =  (A+1.f) + (A-1.f)*cw + 2.f*sA*alpha;
    a1 = -2.f*((A-1.f) + (A+1.f)*cw);
    a2 =  (A+1.f) + (A-1.f)*cw - 2.f*sA*alpha;
  } else if (sec == 4) {     // high shelf
    b0 =  A*((A+1.f) + (A-1.f)*cw + 2.f*sA*alpha);
    b1 = -2.f*A*((A-1.f) + (A+1.f)*cw);
    b2 =  A*((A+1.f) + (A-1.f)*cw - 2.f*sA*alpha);
    a0 =  (A+1.f) - (A-1.f)*cw + 2.f*sA*alpha;
    a1 =  2.f*((A-1.f) - (A+1.f)*cw);
    a2 =  (A+1.f) - (A-1.f)*cw - 2.f*sA*alpha;
  } else {                   // peaking
    b0 = 1.f + alpha*A; b1 = -2.f*cw; b2 = 1.f - alpha*A;
    a0 = 1.f + alpha/A; a1 = -2.f*cw; a2 = 1.f - alpha/A;
  }
  float inv = 1.0f / a0;
  float* o = sosOut + (size_t)(b*5 + sec) * 6;
  o[0]=b0*inv; o[1]=b1*inv; o[2]=b2*inv; o[3]=1.0f; o[4]=a1*inv; o[5]=a2*inv;
}

// --------- Kernel 2: build per-batch block matrices T, D, Bblk, Ablk, Ablk^256 -------
__global__ void eq_build_kernel(const float* __restrict__ sos, float* __restrict__ ws) {
  int b = blockIdx.x * blockDim.x + threadIdx.x;
  if (b >= NBATCH) return;
  float cb0[5],cb1[5],cb2[5],ca1[5],ca2[5];
  for (int k=0;k<5;k++) {
    const float* s = sos + (size_t)(b*5+k)*6;
    cb0[k]=s[0]; cb1[k]=s[1]; cb2[k]=s[2]; ca1[k]=s[4]; ca2[k]=s[5];
  }
  float* M  = ws + MAT_OFF + (size_t)b * MATS_PER_BATCH * 256;
  float* Tm = M;        float* Dm = M + 256;  float* Bm = M + 512;
  float* Am = M + 768;  float* A2 = M + 1024;
  for (int i = 0; i < MATS_PER_BATCH*256; ++i) M[i] = 0.f;

  // state components (z1_k, z2_k)*5 as linear functionals over [x0..x15, s0_0..s0_9]
  float S[10][26]; float cur[26]; float yv[26];
  for (int c=0;c<10;c++) for (int i=0;i<26;i++) S[c][i] = (i == 16+c) ? 1.f : 0.f;

  for (int j=0;j<16;j++) {
    for (int i=0;i<26;i++) cur[i] = (i==j) ? 1.f : 0.f;
    for (int k=0;k<5;k++) {
      for (int i=0;i<26;i++) yv[i] = cb0[k]*cur[i] + S[2*k][i];
      for (int i=0;i<26;i++) {
        float z1n = cb1[k]*cur[i] - ca1[k]*yv[i] + S[2*k+1][i];
        float z2n = cb2[k]*cur[i] - ca2[k]*yv[i];
        S[2*k][i] = z1n; S[2*k+1][i] = z2n;
      }
      for (int i=0;i<26;i++) cur[i] = yv[i];
    }
    for (int i=0;i<16;i++) Tm[j*16+i] = cur[i];
    for (int i=0;i<10;i++) Dm[j*16+i] = cur[16+i];
  }
  for (int c=0;c<10;c++) {
    for (int i=0;i<16;i++) Bm[c*16+i] = S[c][i];
    for (int i=0;i<10;i++) Am[c*16+i] = S[c][16+i];
  }
  // Ablk^256 by repeated squaring (8x) of the 10x10
  float P[10][10], Q[10][10];
  for (int r=0;r<10;r++) for (int c=0;c<10;c++) P[r][c] = S[r][16+c];
  for (int it=0; it<8; ++it) {
    for (int r=0;r<10;r++) for (int c=0;c<10;c++) {
      float acc = 0.f;
      for (int k=0;k<10;k++) acc = fmaf(P[r][k], P[k][c], acc);
      Q[r][c] = acc;
    }
    for (int r=0;r<10;r++) for (int c=0;c<10;c++) P[r][c] = Q[r][c];
  }
  for (int r=0;r<10;r++) for (int c=0;c<10;c++) A2[r*16+c] = P[r][c];
}

// ---- Kernel 3: per-block zero-state end-state contributions E = Bblk @ X via WMMA ---
__global__ void eq_sx_kernel(const float* __restrict__ x, float* __restrict__ ws) {
  int g = blockIdx.x, b = blockIdx.y, l = threadIdx.x;
  int n = l & 15, half = l >> 4;
  const float* Bm = ws + MAT_OFF + (size_t)b*MATS_PER_BATCH*256 + 512;
  const float* ar = Bm + n*16 + 2*half;          // A operand: row M=l&15, K=4c+2*half
  v2f a0v = *(const v2f*)(ar+0),  a1v = *(const v2f*)(ar+4);
  v2f a2v = *(const v2f*)(ar+8),  a3v = *(const v2f*)(ar+12);
  const float* xp = x + (size_t)b*T_LEN + (size_t)g*(GRP*BLK) + n*BLK + 2*half;
  v2f b0v = *(const v2f*)(xp+0),  b1v = *(const v2f*)(xp+4);
  v2f b2v = *(const v2f*)(xp+8),  b3v = *(const v2f*)(xp+12);
  v8f acc = {};
  acc = wmma4(a0v, b0v, acc);
  acc = wmma4(a1v, b1v, acc);
  acc = wmma4(a2v, b2v, acc);
  acc = wmma4(a3v, b3v, acc);
  float* out = ws + SX_OFF + (size_t)(b*NGROUPS + g) * 256;
  int mbase = half * 8;
#pragma unroll
  for (int r=0;r<8;r++) out[(mbase+r)*16 + n] = acc[r];
}

// --------- Kernel 4a: per-chunk zero-state reduction f_c (wave-parallel matvec) ------
__global__ void eq_chunk_reduce_kernel(float* __restrict__ ws) {
  int chunk = blockIdx.x, b = blockIdx.y, r = threadIdx.x;
  const float* Am = ws + MAT_OFF + (size_t)b*MATS_PER_BATCH*256 + 768;
  float Arow[10];
#pragma unroll
  for (int c=0;c<10;c++) Arow[c] = (r < 10) ? Am[r*16+c] : 0.f;
  const float* sx = ws + SX_OFF + (size_t)b*NGROUPS*256;
  float s = 0.f;
  int gbase = chunk * GROUPS_PER_CHUNK;
  for (int blk=0; blk<CHUNK_BLOCKS; ++blk) {
    int g = gbase + (blk >> 4), n = blk & 15;
    float e = (r < 10) ? sx[(size_t)g*256 + r*16 + n] : 0.f;
    float acc = e;
#pragma unroll
    for (int c=0;c<10;c++) acc = fmaf(Arow[c], __shfl(s, c, 32), acc);
    s = acc;
  }
  if (r < 10) ws[CHUNKF_OFF + ((size_t)b*NCHUNKS + chunk)*16 + r] = s;
}

// -------- Kernel 4b: 32-step scan across chunks with Ablk^256 (one lane/batch) -------
__global__ void eq_chunk_scan_kernel(float* __restrict__ ws) {
  int b = threadIdx.x;
  if (b >= NBATCH) return;
  const float* A256 = ws + MAT_OFF + (size_t)b*MATS_PER_BATCH*256 + 1024;
  float s[10];
  for (int r=0;r<10;r++) s[r] = 0.f;
  for (int c=0;c<NCHUNKS;c++) {
    float* so = ws + CHUNKS_OFF + ((size_t)b*NCHUNKS + c)*16;
    for (int r=0;r<10;r++) so[r] = s[r];
    const float* f = ws + CHUNKF_OFF + ((size_t)b*NCHUNKS + c)*16;
    float ns[10];
    for (int r=0;r<10;r++) {
      float acc = f[r];
      for (int k=0;k<10;k++) acc = fmaf(A256[r*16+k], s[k], acc);
      ns[r] = acc;
    }
    for (int r=0;r<10;r++) s[r] = ns[r];
  }
}

// ---- Kernel 5: final outputs y = T@X + D@S via WMMA, state regen inside chunk -------
__global__ void eq_output_kernel(const float* __restrict__ x, const float* __restrict__ ws,
                                 float* __restrict__ y) {
  __shared__ float st[256];           // state tile: [comp 0..15][block n 0..15]
  int chunk = blockIdx.x, b = blockIdx.y, l = threadIdx.x;
  int n = l & 15, half = l >> 4, mbase = half * 8;
  const float* M  = ws + MAT_OFF + (size_t)b*MATS_PER_BATCH*256;
  const float* Tm = M;  const float* Dm = M + 256;  const float* Am = M + 768;
  const float* tr = Tm + n*16 + 2*half;
  v2f ta0 = *(const v2f*)(tr+0), ta1 = *(const v2f*)(tr+4);
  v2f ta2 = *(const v2f*)(tr+8), ta3 = *(const v2f*)(tr+12);
  const float* dr = Dm + n*16 + 2*half;
  v2f da0 = *(const v2f*)(dr+0), da1 = *(const v2f*)(dr+4);
  v2f da2 = *(const v2f*)(dr+8), da3 = *(const v2f*)(dr+12);
  float Arow[10];
#pragma unroll
  for (int c=0;c<10;c++) Arow[c] = (l < 10) ? Am[l*16+c] : 0.f;
  float s = (l < 10) ? ws[CHUNKS_OFF + ((size_t)b*NCHUNKS + chunk)*16 + l] : 0.f;
  const float* sx = ws + SX_OFF + (size_t)b*NGROUPS*256;
  // zero padding rows 10..15 of the state tile (D columns 10..15 are zero, but avoid NaN*0)
  for (int i = l; i < 96; i += 32) st[160 + i] = 0.f;
  __syncthreads();

  for (int gg=0; gg<GROUPS_PER_CHUNK; ++gg) {
    int g = chunk*GROUPS_PER_CHUNK + gg;
    for (int nn=0; nn<16; ++nn) {           // record block-start states, then advance
      if (l < 10) st[l*16 + nn] = s;
      float e = (l < 10) ? sx[(size_t)g*256 + l*16 + nn] : 0.f;
      float acc = e;
#pragma unroll
      for (int c=0;c<10;c++) acc = fmaf(Arow[c], __shfl(s, c, 32), acc);
      s = acc;
    }
    __syncthreads();
    const float* xp = x + (size_t)b*T_LEN + (size_t)g*256 + n*16 + 2*half;
    v2f xb0 = *(const v2f*)(xp+0), xb1 = *(const v2f*)(xp+4);
    v2f xb2 = *(const v2f*)(xp+8), xb3 = *(const v2f*)(xp+12);
    v8f acc = {};
    acc = wmma4(ta0, xb0, acc);
    acc = wmma4(ta1, xb1, acc);
    acc = wmma4(ta2, xb2, acc);
    acc = wmma4(ta3, xb3, acc);
    int kb = 2*half;
    v2f sb0 = { st[(kb+ 0)*16+n], st[(kb+ 1)*16+n] };
    v2f sb1 = { st[(kb+ 4)*16+n], st[(kb+ 5)*16+n] };
    v2f sb2 = { st[(kb+ 8)*16+n], st[(kb+ 9)*16+n] };
    v2f sb3 = { st[(kb+12)*16+n], st[(kb+13)*16+n] };
    acc = wmma4(da0, sb0, acc);
    acc = wmma4(da1, sb1, acc);
    acc = wmma4(da2, sb2, acc);
    acc = wmma4(da3, sb3, acc);
    float* yp = y + (size_t)b*T_LEN + (size_t)g*256 + n*16 + mbase;
    yp[0]=acc[0]; yp[1]=acc[1]; yp[2]=acc[2]; yp[3]=acc[3];
    yp[4]=acc[4]; yp[5]=acc[5]; yp[6]=acc[6]; yp[7]=acc[7];
    __syncthreads();
  }
}

extern "C" void kernel_launch(void* const* d_in, const int* in_sizes, int n_in,
                              void* d_out, int out_size, void* d_ws, size_t ws_size,
                              hipStream_t stream) {
  const float* x  = (const float*)d_in[0];
  const float* cp = (const float*)d_in[1];
  float* yout = (float*)d_out;
  float* sos  = (float*)d_out + (size_t)NBATCH * T_LEN;   // second output, right after y
  float* ws   = (float*)d_ws;

  eq_sos_kernel  <<<dim3(1), dim3(192), 0, stream>>>(cp, sos);
  eq_build_kernel<<<dim3(1), dim3(32),  0, stream>>>(sos, ws);
  eq_sx_kernel   <<<dim3(NGROUPS, NBATCH), dim3(32), 0, stream>>>(x, ws);
  eq_chunk_reduce_kernel<<<dim3(NCHUNKS, NBATCH), dim3(32), 0, stream>>>(ws);
  eq_chunk_scan_kernel  <<<dim3(1), dim3(32), 0, stream>>>(ws);
  eq_output_kernel      <<<dim3(NCHUNKS, NBATCH), dim3(32), 0, stream>>>(x, ws, yout);
}